// NeRFRenderer_19628000543194
// MI455X (gfx1250) — compile-verified
//
#include <hip/hip_runtime.h>
#include <hip/hip_bf16.h>
#include <stdint.h>

#define N_STEPS         128
#define WAVES_PER_BLOCK 8
#define RPW             4        // rays per wave: software-pipeline depth
#define BOUND           1.0f
#define MIN_NEAR        0.2f
#define T_THRESH        1e-4f

#if __has_builtin(__builtin_amdgcn_global_load_async_to_lds_b128) && __has_builtin(__builtin_amdgcn_s_wait_asynccnt)
#define HAVE_ASYNC 1
#endif

#define AS1 __attribute__((address_space(1)))
#define AS3 __attribute__((address_space(3)))

typedef int v4i __attribute__((ext_vector_type(4)));

// Stage 16 bytes global -> LDS. Async DMA (ASYNCcnt) on CDNA5, sync fallback otherwise.
__device__ __forceinline__ void copy16(const float* g, float* l) {
#if defined(HAVE_ASYNC)
    __builtin_amdgcn_global_load_async_to_lds_b128((AS1 v4i*)g, (AS3 v4i*)l, 0, 0);
#else
    *(float4*)l = *(const float4*)g;
#endif
}

#if defined(HAVE_ASYNC)
#define ASYNC_WAIT(n) do { __builtin_amdgcn_s_wait_asynccnt(n); asm volatile("" ::: "memory"); } while (0)
#else
#define ASYNC_WAIT(n) do { asm volatile("" ::: "memory"); } while (0)
#endif

// Issue the 4 coalesced b128 async loads that stage one ray (512B sigma + 1536B rgb)
__device__ __forceinline__ void stage_ray(const float* __restrict__ sigmas,
                                          const float* __restrict__ rgbs,
                                          long r, int lane, float* buf) {
    const float* sg = sigmas + r * (long)N_STEPS;
    const float* rg = rgbs   + r * (long)(3 * N_STEPS);
    copy16(sg +            lane * 4, buf +            lane * 4);   // sigma[0..127]
    copy16(rg +       0  + lane * 4, buf + 128      + lane * 4);   // rgb flat [0..127]
    copy16(rg +     128  + lane * 4, buf + 256      + lane * 4);   // rgb flat [128..255]
    copy16(rg +     256  + lane * 4, buf + 384      + lane * 4);   // rgb flat [256..383]
}

__global__ __launch_bounds__(256) void nerf_composite_kernel(
    const float* __restrict__ rays_o, const float* __restrict__ rays_d,
    const float* __restrict__ sigmas, const float* __restrict__ rgbs,
    float* __restrict__ out, int nrays)
{
    // per-wave double buffer: 2 x 512 floats (sigma 128 + rgb 384)
    __shared__ __align__(16) float lds[WAVES_PER_BLOCK * 2 * 512];

    const int lane = threadIdx.x & 31;
    const int wave = threadIdx.x >> 5;
    float* wbuf = &lds[wave * 1024];

    const long waveId = (long)blockIdx.x * WAVES_PER_BLOCK + wave;
    const long base   = waveId * (long)RPW;
    const long N      = nrays;

    // prologue: stage ray 0 into buffer 0
    if (base < N) stage_ray(sigmas, rgbs, base, lane, wbuf);

#pragma unroll
    for (int i = 0; i < RPW; ++i) {
        const long r = base + i;

        // issue next ray's DMA before consuming current (latency hiding)
        bool issued_next = false;
        if (i + 1 < RPW) {
            const long rn = base + i + 1;
            if (rn < N) {
                stage_ray(sigmas, rgbs, rn, lane, wbuf + ((i + 1) & 1) * 512);
                issued_next = true;
            }
        }
        if (issued_next) { ASYNC_WAIT(4); }   // in-order: current ray's 4 loads done
        else             { ASYNC_WAIT(0); }

        if (r >= N) continue;
        const float* buf = wbuf + (i & 1) * 512;

        // ---- near/far from AABB slab test (lane-redundant, broadcast loads) ----
        float ox = rays_o[r * 3 + 0], oy = rays_o[r * 3 + 1], oz = rays_o[r * 3 + 2];
        float dx = rays_d[r * 3 + 0], dy = rays_d[r * 3 + 1], dz = rays_d[r * 3 + 2];
        dx = (fabsf(dx) < 1e-8f) ? 1e-8f : dx;
        dy = (fabsf(dy) < 1e-8f) ? 1e-8f : dy;
        dz = (fabsf(dz) < 1e-8f) ? 1e-8f : dz;
        float t1x = (-BOUND - ox) / dx, t2x = (BOUND - ox) / dx;
        float t1y = (-BOUND - oy) / dy, t2y = (BOUND - oy) / dy;
        float t1z = (-BOUND - oz) / dz, t2z = (BOUND - oz) / dz;
        float nearv = fmaxf(fmaxf(fminf(t1x, t2x), fminf(t1y, t2y)), fminf(t1z, t2z));
        float farv  = fminf(fminf(fmaxf(t1x, t2x), fmaxf(t1y, t2y)), fmaxf(t1z, t2z));
        nearv = fmaxf(nearv, MIN_NEAR);
        farv  = fmaxf(farv, nearv + 1e-4f);
        const float delta = (farv - nearv) * (1.0f / (float)N_STEPS);

        // ---- composite: wave32 exclusive scan of sigma*delta, T = exp(-cumsum) ----
        float carry = 0.0f;                 // sum of sigma*delta over previous chunks
        float wsum = 0.0f, dep = 0.0f, c0 = 0.0f, c1 = 0.0f, c2 = 0.0f;
#pragma unroll
        for (int ch = 0; ch < N_STEPS / 32; ++ch) {
            const int s  = ch * 32 + lane;
            const float x = buf[s] * delta;          // sigma * delta (DENSITY_SCALE=1)

            // inclusive Hillis-Steele scan across the wave
            float incl = x;
#pragma unroll
            for (int off = 1; off < 32; off <<= 1) {
                float up = __shfl_up(incl, (unsigned)off, 32);
                if (lane >= off) incl += up;
            }
            const float excl  = incl - x;
            const float T     = __expf(-(carry + excl));   // exclusive transmittance
            const float alpha = 1.0f - __expf(-x);
            const float w     = (T > T_THRESH) ? alpha * T : 0.0f;
            const float tmid  = nearv + ((float)s + 0.5f) * delta;

            wsum += w;
            dep  += w * tmid;
            const float* rp = buf + 128 + s * 3;     // bank-conflict-free (stride 3)
            c0 += w * rp[0];
            c1 += w * rp[1];
            c2 += w * rp[2];

            carry += __shfl(incl, 31, 32);           // broadcast chunk total
        }

        // ---- wave reduction over 32 lanes ----
#pragma unroll
        for (int off = 16; off > 0; off >>= 1) {
            wsum += __shfl_xor(wsum, off, 32);
            dep  += __shfl_xor(dep,  off, 32);
            c0   += __shfl_xor(c0,   off, 32);
            c1   += __shfl_xor(c1,   off, 32);
            c2   += __shfl_xor(c2,   off, 32);
        }

        if (lane == 0) {
            const float bg = 1.0f - wsum;            // BG_COLOR = 1.0
            out[r * 3 + 0] = c0 + bg;
            out[r * 3 + 1] = c1 + bg;
            out[r * 3 + 2] = c2 + bg;
            out[3 * N + r] = dep;
            out[4 * N + r] = fmaxf(dep - nearv, 0.0f) / (farv - nearv);
        }
    }
}

extern "C" void kernel_launch(void* const* d_in, const int* in_sizes, int n_in,
                              void* d_out, int out_size, void* d_ws, size_t ws_size,
                              hipStream_t stream) {
    const float* rays_o = (const float*)d_in[0];
    const float* rays_d = (const float*)d_in[1];
    const float* sigmas = (const float*)d_in[2];
    const float* rgbs   = (const float*)d_in[3];
    float* out = (float*)d_out;

    const int n = in_sizes[0] / 3;                       // N_RAYS
    const int raysPerBlock = WAVES_PER_BLOCK * RPW;      // 32
    const int grid = (n + raysPerBlock - 1) / raysPerBlock;
    nerf_composite_kernel<<<grid, 256, 0, stream>>>(rays_o, rays_d, sigmas, rgbs, out, n);
}